// KAN_PINN_12979391169010
// MI455X (gfx1250) — compile-verified
//
#include <hip/hip_runtime.h>

// KAN-PINN forward, MI455X (gfx1250, wave32).
// Compute-bound on the transcendental pipe: ~537M tanh + ~1.1G FMA vs ~15MB
// of traffic (irrelevant at 23.3 TB/s). Strategy:
//  - native v_tanh_f32 (gfx1250 TRANS op)
//  - wave-uniform j (readfirstlane) -> weights via scalar loads / SGPRs
//  - activations stored transposed hT[i][n] -> coalesced lane loads
//  - 4 points per lane for TRANS-latency hiding
//  - final (N x 64) @ (64 x 1) projection via V_WMMA_F32_16X16X4_F32,
//    B built from readfirstlane'd scalar weights * column mask (no branches)

#define NPTS 8192
#define HDIM 64
#define UHD  8
#define NP   4   // points per lane in the KAN layer kernels

typedef __attribute__((ext_vector_type(2))) float v2f;
typedef __attribute__((ext_vector_type(8))) float v8f;

__device__ __forceinline__ float fast_tanh(float x) {
#if __has_builtin(__builtin_amdgcn_tanhf)
  return __builtin_amdgcn_tanhf(x);
#else
  float r;
  asm("v_tanh_f32 %0, %1" : "=v"(r) : "v"(x));
  return r;
#endif
}

// Convergent scalar-ization: cannot be sunk into divergent control flow,
// so the feeding (uniform-address) load stays a plain s_load.
__device__ __forceinline__ float rfl_f(float x) {
  return __int_as_float(__builtin_amdgcn_readfirstlane(__float_as_int(x)));
}

// ---------------- Layer 0: d_in = 2 (inputs x, t) ----------------
__global__ __launch_bounds__(256) void kan_layer0_kernel(
    const float* __restrict__ x, const float* __restrict__ t,
    const float* __restrict__ W1, const float* __restrict__ B1,
    const float* __restrict__ W2, const float* __restrict__ B2,
    float* __restrict__ outT)   // [HDIM][NPTS]
{
  const int lane = threadIdx.x & 31;
  int j = blockIdx.y * 8 + (threadIdx.x >> 5);
  j = __builtin_amdgcn_readfirstlane(j);          // force SGPR -> s_load weights
  const int n0 = blockIdx.x * (32 * NP) + lane;

  float hv[2][NP];
#pragma unroll
  for (int k = 0; k < NP; ++k) {
    hv[0][k] = x[n0 + 32 * k];
    hv[1][k] = t[n0 + 32 * k];
  }

  float acc[NP] = {0.f, 0.f, 0.f, 0.f};
  float sB2 = 0.f;
#pragma unroll
  for (int i = 0; i < 2; ++i) {
    const float* w1 = W1 + (j * 2 + i) * UHD;
    const float* b1 = B1 + (j * 2 + i) * UHD;
    const float* w2 = W2 + (j * 2 + i) * UHD;
    sB2 += B2[j * 2 + i];
#pragma unroll
    for (int u = 0; u < UHD; ++u) {
      const float a = w1[u], b = b1[u], c = w2[u];
#pragma unroll
      for (int k = 0; k < NP; ++k)
        acc[k] = fmaf(c, fast_tanh(fmaf(hv[i][k], a, b)), acc[k]);
    }
  }
#pragma unroll
  for (int k = 0; k < NP; ++k)
    outT[j * NPTS + n0 + 32 * k] = fast_tanh(acc[k] + sB2);
}

// ---------------- Mid layers: d_in = 64 ----------------
__global__ __launch_bounds__(256) void kan_layer64_kernel(
    const float* __restrict__ hT,   // [HDIM][NPTS]
    const float* __restrict__ W1, const float* __restrict__ B1,
    const float* __restrict__ W2, const float* __restrict__ B2,
    float* __restrict__ outT)       // [HDIM][NPTS]
{
  const int lane = threadIdx.x & 31;
  int j = blockIdx.y * 8 + (threadIdx.x >> 5);
  j = __builtin_amdgcn_readfirstlane(j);          // weights wave-uniform
  const int n0 = blockIdx.x * (32 * NP) + lane;

  float acc[NP] = {0.f, 0.f, 0.f, 0.f};
  float sB2 = 0.f;

#pragma unroll 2
  for (int i = 0; i < HDIM; ++i) {
    float hv[NP];
#pragma unroll
    for (int k = 0; k < NP; ++k) hv[k] = hT[i * NPTS + n0 + 32 * k];

    const float* w1 = W1 + (j * HDIM + i) * UHD;
    const float* b1 = B1 + (j * HDIM + i) * UHD;
    const float* w2 = W2 + (j * HDIM + i) * UHD;
    sB2 += B2[j * HDIM + i];

#pragma unroll
    for (int u = 0; u < UHD; ++u) {
      const float a = w1[u], b = b1[u], c = w2[u];
#pragma unroll
      for (int k = 0; k < NP; ++k)
        acc[k] = fmaf(c, fast_tanh(fmaf(hv[k], a, b)), acc[k]);
    }
  }
#pragma unroll
  for (int k = 0; k < NP; ++k)
    outT[j * NPTS + n0 + 32 * k] = fast_tanh(acc[k] + sB2);
}

// ---------------- Output projection: u = h3 @ W_out^T + b_out ----------------
// Per wave: one 16-point tile. C(16x16) = sum_k A(16x4) * B(4x16), where
// A[m,k] = h3[n0+m, k] (from transposed hT) and B has W_out in column 0 only.
// f32 16x16x4 A/B layout: lanes 0-15 hold (K, K+1) for K base 0, lanes 16-31
// for K base 2; C/D column 0 lives in lanes 0 (M=0..7) and 16 (M=8..15).
__global__ __launch_bounds__(256) void kan_out_proj_kernel(
    const float* __restrict__ hT,    // [HDIM][NPTS]
    const float* __restrict__ Wout,  // [HDIM]
    const float* __restrict__ bout,  // [1]
    float* __restrict__ out)         // [NPTS]
{
  const int lane = threadIdx.x & 31;
  const int wave = threadIdx.x >> 5;
  const int n0 = (blockIdx.x * 8 + wave) * 16;
  const bool lo = (lane < 16);                 // K half selector
  const int m = lane & 15;                     // row (point) index within tile
  const int kh = lo ? 0 : 2;
  const float cmask = (m == 0) ? 1.f : 0.f;    // column-0 mask (per-lane VGPR)

#if __has_builtin(__builtin_amdgcn_wmma_f32_16x16x4_f32)
  v8f c = {0.f, 0.f, 0.f, 0.f, 0.f, 0.f, 0.f, 0.f};
#pragma unroll
  for (int k0 = 0; k0 < HDIM; k0 += 4) {
    // A tile: per-lane coalesced loads from transposed activations.
    v2f a;
    a.x = hT[(k0 + kh + 0) * NPTS + n0 + m];
    a.y = hT[(k0 + kh + 1) * NPTS + n0 + m];
    // Weights: uniform s_loads pinned to SGPRs via convergent readfirstlane;
    // column masking via multiply -> pure VALU, nothing sinkable.
    const float w0 = rfl_f(Wout[k0 + 0]);
    const float w1 = rfl_f(Wout[k0 + 1]);
    const float w2v = rfl_f(Wout[k0 + 2]);
    const float w3 = rfl_f(Wout[k0 + 3]);
    v2f b;
    b.x = (lo ? w0 : w2v) * cmask;
    b.y = (lo ? w1 : w3) * cmask;
    c = __builtin_amdgcn_wmma_f32_16x16x4_f32(
        /*neg_a=*/false, a, /*neg_b=*/false, b,
        /*c_mod=*/(short)0, c, /*reuse_a=*/false, /*reuse_b=*/false);
  }
  const float bo = rfl_f(bout[0]);
  if (m == 0) {
    const int base = n0 + (lo ? 0 : 8);
#pragma unroll
    for (int r = 0; r < 8; ++r) out[base + r] = c[r] + bo;
  }
#else
  // VALU fallback (also tells us via histogram if the builtin was absent)
  if (lane < 16) {
    float acc = bout[0];
    for (int k = 0; k < HDIM; ++k) acc = fmaf(Wout[k], hT[k * NPTS + n0 + m], acc);
    out[n0 + m] = acc;
  }
#endif
}

extern "C" void kernel_launch(void* const* d_in, const int* in_sizes, int n_in,
                              void* d_out, int out_size, void* d_ws, size_t ws_size,
                              hipStream_t stream) {
  const float* x    = (const float*)d_in[0];
  const float* t    = (const float*)d_in[1];
  const float* W1_0 = (const float*)d_in[2];
  const float* B1_0 = (const float*)d_in[3];
  const float* W2_0 = (const float*)d_in[4];
  const float* B2_0 = (const float*)d_in[5];
  const float* W1_1 = (const float*)d_in[6];
  const float* B1_1 = (const float*)d_in[7];
  const float* W2_1 = (const float*)d_in[8];
  const float* B2_1 = (const float*)d_in[9];
  const float* W1_2 = (const float*)d_in[10];
  const float* B1_2 = (const float*)d_in[11];
  const float* W2_2 = (const float*)d_in[12];
  const float* B2_2 = (const float*)d_in[13];
  const float* Wout = (const float*)d_in[14];
  const float* bout = (const float*)d_in[15];

  float* bufA = (float*)d_ws;            // [HDIM][NPTS] transposed activations
  float* bufB = bufA + (size_t)NPTS * HDIM;

  const dim3 blk(256);
  const dim3 gl(NPTS / (32 * NP), HDIM / 8);   // (64, 8): 4096 waves

  kan_layer0_kernel<<<gl, blk, 0, stream>>>(x, t, W1_0, B1_0, W2_0, B2_0, bufA);
  kan_layer64_kernel<<<gl, blk, 0, stream>>>(bufA, W1_1, B1_1, W2_1, B2_1, bufB);
  kan_layer64_kernel<<<gl, blk, 0, stream>>>(bufB, W1_2, B1_2, W2_2, B2_2, bufA);
  kan_out_proj_kernel<<<dim3(NPTS / 128), blk, 0, stream>>>(bufA, Wout, bout,
                                                            (float*)d_out);
}